// Yolo_postprocessor_68874095559053
// MI455X (gfx1250) — compile-verified
//
#include <hip/hip_runtime.h>
#include <hip/hip_bf16.h>
#include <stdint.h>

#define N_BOXES 8192
#define N_COLS  85
#define CONF_THRES 0.25f
#define IOU_THRES  0.45f
#define NEGV (-1000000000.0f)
#define NPRE 2048              // rows with precomputed suppression bitmask
#define SEGS 512               // 8192/16 u16 segments per mask row

typedef __attribute__((ext_vector_type(2))) float v2f;
typedef __attribute__((ext_vector_type(8))) float v8f;

// ---- workspace layout (bytes) ----
#define OFF_BOXES   0u         // float4 * 8192   (decoded boxes, original order)
#define OFF_CONF    131072u    // float  * 8192
#define OFF_KEYS    163840u    // u64    * 8192   (sort keys)
#define OFF_SIDX    229376u    // int    * 8192   (sorted -> original index)
#define OFF_SBOX    262144u    // float4 * 8192   (sorted boxes)
#define OFF_SAREA   393216u    // float  * 8192
#define OFF_SVALID  425984u    // int    * 8192
#define OFF_KEEPO   458752u    // int    * 8192   (keep, original order)
#define OFF_MASK    491520u    // u16 * NPRE * SEGS = 2 MB

// ---------------- decode ----------------
__global__ void k_decode(const float* __restrict__ x,
                         const int* __restrict__ piw, const int* __restrict__ pih,
                         float4* __restrict__ boxes, float* __restrict__ conf,
                         unsigned long long* __restrict__ keys) {
  int b = blockIdx.x * blockDim.x + threadIdx.x;
  if (b >= N_BOXES) return;
  int iwi = *piw, ihi = *pih;
  float iw = (float)iwi, ih = (float)ihi;
  const float* row = x + (size_t)b * N_COLS;
  float cf = row[4];
  float best = row[5]; int bi = 0;
  #pragma unroll 8
  for (int c = 1; c < 80; ++c) { float v = row[5 + c]; if (v > best) { best = v; bi = c; } }
  // replicate reference's height/width swap
  float bx = row[0] * ih, by = row[1] * iw, bw = row[2] * ih, bh = row[3] * iw;
  bool valid = (cf > CONF_THRES) && (bi == 0) && ((bw > 0.f) || (bh > 0.f));
  float hw = floorf(bw * 0.5f), hh = floorf(bh * 0.5f);
  int x1 = max((int)(bx - hw), 0);
  int y1 = max((int)(by - hh), 0);
  int x2 = min((int)(bx + hw), iwi);
  int y2 = min((int)(by + hh), ihi);
  boxes[b] = make_float4((float)x1, (float)y1, (float)x2, (float)y2);
  conf[b] = cf;
  float s = valid ? cf : NEGV;
  unsigned u = __float_as_uint(s);
  u = (u & 0x80000000u) ? ~u : (u | 0x80000000u);     // order-preserving key
  keys[b] = ((unsigned long long)u << 32) | (unsigned)(N_BOXES - 1 - b); // stable ties
}

// ---------------- bitonic sort (one workgroup, 64KB LDS) ----------------
__global__ __launch_bounds__(1024) void k_sort(const unsigned long long* __restrict__ keys,
                                               const float4* __restrict__ boxes,
                                               int* __restrict__ sidx, float4* __restrict__ sbox,
                                               float* __restrict__ sarea, int* __restrict__ svalid) {
  __shared__ unsigned long long sk[N_BOXES];
  int tid = threadIdx.x;
  for (int i = tid; i < N_BOXES; i += 1024) sk[i] = keys[i];
  __syncthreads();
  for (int k = 2; k <= N_BOXES; k <<= 1) {
    for (int j = k >> 1; j > 0; j >>= 1) {
      for (int i = tid; i < N_BOXES; i += 1024) {
        int l = i ^ j;
        if (l > i) {
          unsigned long long a = sk[i], b2 = sk[l];
          bool desc = (i & k) == 0;                    // overall descending
          if (desc ? (a < b2) : (a > b2)) { sk[i] = b2; sk[l] = a; }
        }
      }
      __syncthreads();
    }
  }
  for (int i = tid; i < N_BOXES; i += 1024) {
    unsigned long long kk = sk[i];
    int orig = (N_BOXES - 1) - (int)(kk & 0xFFFFFFFFull);
    unsigned u = (unsigned)(kk >> 32);
    float4 bb = boxes[orig];
    sidx[i] = orig;
    sbox[i] = bb;
    sarea[i] = (bb.z - bb.x) * (bb.w - bb.y);
    svalid[i] = (int)(u >> 31);                        // valid scores were positive
  }
}

// ---------------- suppression bitmask: one 16x16 tile per wave32 ----------------
// Tiles whose first row/col is invalid can never affect the output (sorted order:
// valid boxes first; scan breaks at first invalid row; keep = svalid && !removed),
// so those waves exit immediately. No block barrier -> early return is safe.
__global__ __launch_bounds__(256) void k_mask(const float4* __restrict__ sbox,
                                              const float* __restrict__ sarea,
                                              const int* __restrict__ svalid,
                                              unsigned short* __restrict__ mask) {
  int wid  = threadIdx.x >> 5;
  int lane = threadIdx.x & 31;
  int tile = blockIdx.x * 8 + wid;
  int rt = tile / SEGS;                 // row tile 0..NPRE/16-1
  int ct = tile % SEGS;                 // col tile 0..511
  if (svalid[rt * 16] == 0 || svalid[ct * 16] == 0) return;   // wave-uniform
  int n = lane & 15;
  bool lo = lane < 16;
  float4 cb = sbox[ct * 16 + n];
  float  ca = sarea[ct * 16 + n];
  float  ra = sarea[rt * 16 + n];

  // Outer-sum of areas via matrix unit: S[m][n] = area_row[m] + area_col[n]
  // A(16x4): col0 = area_row, col1 = 1 ; B(4x16): row0 = 1, row1 = area_col
  v8f S;
#if __has_builtin(__builtin_amdgcn_wmma_f32_16x16x4_f32)
  {
    v2f A, B;
    A.x = lo ? ra  : 0.f;  A.y = lo ? 1.f : 0.f;   // lanes 16-31 supply K2,K3 = 0
    B.x = lo ? 1.f : 0.f;  B.y = lo ? ca  : 0.f;
    v8f C = {0.f, 0.f, 0.f, 0.f, 0.f, 0.f, 0.f, 0.f};
    S = __builtin_amdgcn_wmma_f32_16x16x4_f32(false, A, false, B, (short)0, C, false, false);
  }
#else
  #pragma unroll
  for (int v = 0; v < 8; ++v) S[v] = sarea[rt * 16 + v + (lo ? 0 : 8)] + ca;
#endif

  const float t1 = 1.0f + IOU_THRES;
  #pragma unroll
  for (int v = 0; v < 8; ++v) {
    int m = v + (lo ? 0 : 8);                         // matches WMMA C/D VGPR layout
    float4 rb = sbox[rt * 16 + m];                    // 2 uniform addrs/load, WGP$-hot
    float ix1 = fmaxf(rb.x, cb.x), iy1 = fmaxf(rb.y, cb.y);
    float ix2 = fminf(rb.z, cb.z), iy2 = fminf(rb.w, cb.w);
    float inter = fmaxf(ix2 - ix1, 0.f) * fmaxf(iy2 - iy1, 0.f);
    // inter/union > T  <=>  inter*(1+T) > T*(a_i + a_j)   (union >= inter >= 0)
    bool sup = inter * t1 > IOU_THRES * S[v];
#if __has_builtin(__builtin_amdgcn_ballot_w32)
    unsigned bal = __builtin_amdgcn_ballot_w32(sup);
#else
    unsigned bal = (unsigned)__ballot(sup);
#endif
    if (lane == 0) {
      mask[(size_t)(rt * 16 + v) * SEGS + ct]     = (unsigned short)(bal & 0xFFFFu);
      mask[(size_t)(rt * 16 + v + 8) * SEGS + ct] = (unsigned short)(bal >> 16);
    }
  }
}

// ---------------- serial greedy scan over sorted order ----------------
__global__ __launch_bounds__(1024) void k_scan(const int* __restrict__ svalid,
                                               const float4* __restrict__ sbox,
                                               const float* __restrict__ sarea,
                                               const unsigned short* __restrict__ mask,
                                               const int* __restrict__ sidx,
                                               int* __restrict__ keep_orig) {
  __shared__ unsigned removed[N_BOXES / 32];          // 256 words
  int tid = threadIdx.x;
  for (int w = tid; w < N_BOXES / 32; w += 1024) removed[w] = 0u;
  __syncthreads();
  const float t1 = 1.0f + IOU_THRES;
  for (int i = 0; i < N_BOXES; ++i) {
    if (!svalid[i]) break;                            // invalids sort last
    if ((removed[i >> 5] >> (i & 31)) & 1u) continue; // uniform across block
    if (i < NPRE) {
      const unsigned short* rowm = mask + (size_t)i * SEGS;
      // pull next row toward the cache while we consume this one
      if (tid < 8 && i + 1 < NPRE)
        __builtin_prefetch(rowm + SEGS + (size_t)tid * 64, 0, 0);
      for (int s = tid; s < SEGS; s += 1024) {
        int jbase = s * 16;
        if (jbase + 15 <= i) continue;                // whole segment j <= i
        unsigned bits = rowm[s];
        if (jbase <= i) bits &= (0xFFFFu << (i - jbase + 1)); // keep only j > i
        if (bits) atomicOr(&removed[s >> 1], bits << ((s & 1) * 16));
      }
    } else {                                          // rare fallback beyond NPRE
      float4 bi = sbox[i];
      float ai = sarea[i];
      for (int j = i + 1 + tid; j < N_BOXES; j += 1024) {
        float4 bj = sbox[j];
        float ix1 = fmaxf(bi.x, bj.x), iy1 = fmaxf(bi.y, bj.y);
        float ix2 = fminf(bi.z, bj.z), iy2 = fminf(bi.w, bj.w);
        float inter = fmaxf(ix2 - ix1, 0.f) * fmaxf(iy2 - iy1, 0.f);
        if (inter * t1 > IOU_THRES * (ai + sarea[j]))
          atomicOr(&removed[j >> 5], 1u << (j & 31));
      }
    }
    __syncthreads();
  }
  __syncthreads();
  for (int i = tid; i < N_BOXES; i += 1024) {
    bool k = svalid[i] && !((removed[i >> 5] >> (i & 31)) & 1u);
    keep_orig[sidx[i]] = k ? 1 : 0;
  }
}

// ---------------- scatter outputs ----------------
__global__ void k_out(const float4* __restrict__ boxes, const float* __restrict__ conf,
                      const int* __restrict__ keep_orig, float* __restrict__ out) {
  int b = blockIdx.x * blockDim.x + threadIdx.x;
  if (b >= N_BOXES) return;
  int k = keep_orig[b];
  float4 bb = boxes[b];
  out[b * 4 + 0] = k ? bb.x : 0.f;
  out[b * 4 + 1] = k ? bb.y : 0.f;
  out[b * 4 + 2] = k ? bb.z : 0.f;
  out[b * 4 + 3] = k ? bb.w : 0.f;
  out[N_BOXES * 4 + b] = k ? conf[b] : 0.f;
  out[N_BOXES * 5 + b] = k ? 1.f : 0.f;
}

extern "C" void kernel_launch(void* const* d_in, const int* in_sizes, int n_in,
                              void* d_out, int out_size, void* d_ws, size_t ws_size,
                              hipStream_t stream) {
  const float* x  = (const float*)d_in[0];
  const int* piw  = (const int*)d_in[1];
  const int* pih  = (const int*)d_in[2];
  char* ws = (char*)d_ws;
  float4* boxes            = (float4*)(ws + OFF_BOXES);
  float* conf              = (float*)(ws + OFF_CONF);
  unsigned long long* keys = (unsigned long long*)(ws + OFF_KEYS);
  int* sidx                = (int*)(ws + OFF_SIDX);
  float4* sbox             = (float4*)(ws + OFF_SBOX);
  float* sarea             = (float*)(ws + OFF_SAREA);
  int* svalid              = (int*)(ws + OFF_SVALID);
  int* keepo               = (int*)(ws + OFF_KEEPO);
  unsigned short* mask     = (unsigned short*)(ws + OFF_MASK);

  k_decode<<<N_BOXES / 256, 256, 0, stream>>>(x, piw, pih, boxes, conf, keys);
  k_sort<<<1, 1024, 0, stream>>>(keys, boxes, sidx, sbox, sarea, svalid);
  k_mask<<<(NPRE / 16) * SEGS / 8, 256, 0, stream>>>(sbox, sarea, svalid, mask);
  k_scan<<<1, 1024, 0, stream>>>(svalid, sbox, sarea, mask, sidx, keepo);
  k_out<<<N_BOXES / 256, 256, 0, stream>>>(boxes, conf, keepo, (float*)d_out);
}